// RoPE_88974542504622
// MI455X (gfx1250) — compile-verified
//
#include <hip/hip_runtime.h>

// MI455X / gfx1250, wave32. RoPE as block-diagonal WMMA:
// out_tile(16x16) = x_tile(16x16) @ R_block(16x16), R from rope's diagonal band.
// 4x V_WMMA_F32_16X16X4_F32 per tile; HBM-bound (~512MB @ 23.3TB/s ~ 22us floor).
// v2: pinned loop-invariant B fragments + double-buffered, software-pipelined LDS.

typedef __attribute__((ext_vector_type(2))) float v2f;
typedef __attribute__((ext_vector_type(4))) float v4f;
typedef __attribute__((ext_vector_type(8))) float v8f;

#define HIDDEN            4096
#define ROWS_TOTAL        (4 * 4096)   // 16384 sequence rows (B*S)
#define COLS_PER_BLOCK    256          // 8 waves * 2 tiles * 16 cols
#define TILES_PER_WAVE    2
#define ROWTILES_PER_BLK  8
#define LDS_STRIDE        260          // 256 + 4 floats pad: conflict-free A reads

__global__ __launch_bounds__(256)
void rope_wmma_kernel(const float* __restrict__ x,
                      const float* __restrict__ rope,
                      float* __restrict__ out)
{
    __shared__ float xs[2][16 * LDS_STRIDE];   // ping-pong 16x256 slices (2 x 16.25 KB)

    const int tid   = threadIdx.x;
    const int lane  = tid & 31;
    const int wave  = tid >> 5;      // 0..7
    const int n     = lane & 15;     // A: row M / D: col N for this lane
    const int khalf = lane >> 4;     // 0: K{0,1}, 1: K{2,3} halves (ISA 7.12.2 f32 layouts)

    const int colbase = blockIdx.x * COLS_PER_BLOCK;

    // ---- Loop-invariant B fragments (rotation blocks), built once per wave. ----
    // R[2i,2i]=c, R[2i,2i+1]=-s, R[2i+1,2i]=s, R[2i+1,2i+1]=c.
    // Chunk kk covers rows 4kk..4kk+3; this lane's rows: r0=4kk+2*khalf (VGPR0), r0+1
    // (VGPR1) -> pair pr = 2kk+khalf.
    v2f bfrag[TILES_PER_WAVE][4];
#pragma unroll
    for (int t = 0; t < TILES_PER_WAVE; ++t) {
        const int tilecol = colbase + (wave * TILES_PER_WAVE + t) * 16;
#pragma unroll
        for (int kk = 0; kk < 4; ++kk) {
            const int pr = 2 * kk + khalf;
            const int P  = (tilecol >> 1) + pr;               // global rotation-pair index
            const float c = rope[(size_t)(2 * P)     * HIDDEN + 2 * P];  // cos
            const float s = rope[(size_t)(2 * P + 1) * HIDDEN + 2 * P];  // sin
            float b0 = 0.0f, b1 = 0.0f;
            if (n == 2 * pr)     { b0 =  c; b1 = s; }
            if (n == 2 * pr + 1) { b0 = -s; b1 = c; }
            // Opaque-pin: force the selects to resolve ONCE into live VGPRs instead of
            // being rematerialized (v_cmp/v_cndmask chains) inside every row-tile.
            asm("" : "+v"(b0), "+v"(b1));
            bfrag[t][kk] = (v2f){b0, b1};
        }
    }

    const int rowtile0 = blockIdx.y * ROWTILES_PER_BLK;

    // Per-thread slot in the cooperative 16x256 slice copy: fixed column, 4 rows.
    const int rquarter = tid >> 6;      // 0..3
    const int c4       = tid & 63;      // float4 column slot
    const size_t gcol  = (size_t)(colbase + c4 * 4);

    // ---- Prologue: fill buffer 0 with row-tile 0. ----
    v4f stage[4];
#pragma unroll
    for (int i = 0; i < 4; ++i) {
        const int r = 4 * i + rquarter;
        stage[i] = __builtin_nontemporal_load(
            (const v4f*)(x + (size_t)(rowtile0 * 16 + r) * HIDDEN + gcol));
    }
#pragma unroll
    for (int i = 0; i < 4; ++i) {
        const int r = 4 * i + rquarter;
        *(v4f*)(&xs[0][r * LDS_STRIDE + c4 * 4]) = stage[i];
    }
    __syncthreads();

    for (int rt = 0; rt < ROWTILES_PER_BLK; ++rt) {
        const int rowbase = (rowtile0 + rt) * 16;
        const int cur     = rt & 1;

        // ---- Issue next tile's global loads NOW; latency hides under WMMA+stores.
        if (rt + 1 < ROWTILES_PER_BLK) {
#pragma unroll
            for (int i = 0; i < 4; ++i) {
                const int r = 4 * i + rquarter;
                const size_t goff = (size_t)(rowbase + 16 + r) * HIDDEN + gcol;
                stage[i] = __builtin_nontemporal_load((const v4f*)(x + goff));
                if (rt + 2 < ROWTILES_PER_BLK) {
                    // global_prefetch_b8: warm GL2 one more tile ahead.
                    __builtin_prefetch(x + goff + (size_t)16 * HIDDEN, 0, 1);
                }
            }
        }

        // ---- Compute: per wave, 2 column tiles x 4 K-chunks of WMMA f32 16x16x4. ----
#pragma unroll
        for (int t = 0; t < TILES_PER_WAVE; ++t) {
            const int tcol = (wave * TILES_PER_WAVE + t) * 16;   // local col in slice

            v8f acc = {};                                        // C = 0
#pragma unroll
            for (int kk = 0; kk < 4; ++kk) {
                // A fragment: lane holds x[M = n, K = 4kk + 2*khalf + {0,1}]
                const v2f a = *(const v2f*)(&xs[cur][n * LDS_STRIDE + tcol + 4 * kk + 2 * khalf]);
                // 8 args: (neg_a, A, neg_b, B, c_mod, C, reuse_a, reuse_b)
                acc = __builtin_amdgcn_wmma_f32_16x16x4_f32(
                        false, a, false, bfrag[t][kk], (short)0, acc, false, false);
            }

            // ---- Store D: VGPR g -> row (g + 8*khalf), col n; 64B segments, NT. ----
            float* op = out + (size_t)rowbase * HIDDEN + colbase + tcol + n;
#pragma unroll
            for (int g = 0; g < 8; ++g) {
                __builtin_nontemporal_store(acc[g], op + (size_t)(g + 8 * khalf) * HIDDEN);
            }
        }

        // ---- Drain staged next-tile data into the alternate LDS buffer. ----
        if (rt + 1 < ROWTILES_PER_BLK) {
#pragma unroll
            for (int i = 0; i < 4; ++i) {
                const int r = 4 * i + rquarter;
                *(v4f*)(&xs[cur ^ 1][r * LDS_STRIDE + c4 * 4]) = stage[i];
            }
        }
        __syncthreads();   // single barrier per iteration (buffers are disjoint)
    }
}

extern "C" void kernel_launch(void* const* d_in, const int* in_sizes, int n_in,
                              void* d_out, int out_size, void* d_ws, size_t ws_size,
                              hipStream_t stream) {
    const float* x    = (const float*)d_in[0];   // [4,4096,4096] f32
    const float* rope = (const float*)d_in[1];   // [4096,4096]  f32 (block-diagonal)
    float* out        = (float*)d_out;           // [4,4096,4096] f32

    dim3 grid(HIDDEN / COLS_PER_BLOCK,                       // 16 column blocks
              ROWS_TOTAL / (16 * ROWTILES_PER_BLK));         // 128 row blocks
    rope_wmma_kernel<<<grid, 256, 0, stream>>>(x, rope, out);
}